// MyGraphSage_11622181503636
// MI455X (gfx1250) — compile-verified
//
#include <hip/hip_runtime.h>
#include <hip/hip_bf16.h>

typedef __attribute__((ext_vector_type(2))) float v2f;
typedef __attribute__((ext_vector_type(8))) float v8f;

#define DIM     128
#define TILE_M  16
#define LDH     132   // padded row pitch for the A-tile in LDS (bank-conflict avoidance)
#define KSLAB   64    // K rows of W staged per LDS slab (32KB)

// ---------------------------------------------------------------------------
// zero a float buffer
__global__ void zero_f32_kernel(float* __restrict__ p, int n) {
    int i = blockIdx.x * blockDim.x + threadIdx.x;
    if (i < n) p[i] = 0.0f;
}

// ---------------------------------------------------------------------------
// deg[dst[e]] += 1.0f
__global__ void degree_kernel(const int* __restrict__ dst,
                              float* __restrict__ deg, int E) {
    int e = blockIdx.x * blockDim.x + threadIdx.x;
    if (e < E) atomicAdd(&deg[dst[e]], 1.0f);
}

// ---------------------------------------------------------------------------
// acc[dst[e]][:] += x[src[e]][:]   (one wave32 per edge, float4 per lane)
__global__ void scatter_add_kernel(const float* __restrict__ x,
                                   const int* __restrict__ src,
                                   const int* __restrict__ dst,
                                   float* __restrict__ acc, int E) {
    int wid  = (int)((blockIdx.x * blockDim.x + threadIdx.x) >> 5);
    int lane = threadIdx.x & 31;
    if (wid >= E) return;
    size_t s = (size_t)src[wid];
    size_t d = (size_t)dst[wid];
    const float4 v = ((const float4*)(x + s * DIM))[lane];
    float* p = acc + d * DIM + (size_t)lane * 4;
    atomicAdd(p + 0, v.x);
    atomicAdd(p + 1, v.y);
    atomicAdd(p + 2, v.z);
    atomicAdd(p + 3, v.w);
}

// ---------------------------------------------------------------------------
// out = leaky_relu( ((aggr + self) * 1/(deg+1)) @ W + bias )
// One workgroup (8 waves) per 16-row tile. Wave w computes columns [16w,16w+16).
// Uses V_WMMA_F32_16X16X4_F32 with K-loop of 32 steps.
__global__ __launch_bounds__(256)
void sage_gemm_wmma_kernel(const float* __restrict__ aggr,
                           const float* __restrict__ self,
                           const float* __restrict__ deg,
                           const float* __restrict__ W,     // [DIM][DIM] row-major (k,n)
                           const float* __restrict__ bias,  // [DIM]
                           float* __restrict__ out,         // [n_nodes][DIM]
                           int n_nodes) {
    __shared__ float sW[KSLAB * DIM];     // 32 KB K-slab of W
    __shared__ float sH[TILE_M * LDH];    // ~8.4 KB padded input tile [m][k]

    const int tid  = threadIdx.x;     // 0..255
    const int wave = tid >> 5;        // 0..7
    const int lane = tid & 31;
    const int row0 = blockIdx.x * TILE_M;

    // ---- build sH[m][k] = (aggr + self) * 1/(deg+1), 512 float4s / 256 thr ----
    #pragma unroll
    for (int i = 0; i < 2; ++i) {
        int idx = tid + 256 * i;          // 0..511
        int m   = idx >> 5;               // tile row
        int kv  = idx & 31;               // float4 index within row
        int r   = row0 + m;
        if (r >= n_nodes) r = n_nodes - 1;
        float scale = 1.0f / (deg[r] + 1.0f);
        const float4 a = ((const float4*)(aggr + (size_t)r * DIM))[kv];
        const float4 f = ((const float4*)(self + (size_t)r * DIM))[kv];
        float4 h;
        h.x = (a.x + f.x) * scale;
        h.y = (a.y + f.y) * scale;
        h.z = (a.z + f.z) * scale;
        h.w = (a.w + f.w) * scale;
        *(float4*)&sH[m * LDH + 4 * kv] = h;
    }

    const int n0   = wave * 16;
    const int ln   = lane & 15;
    const int koff = (lane < 16) ? 0 : 2;  // K-pair split across half-waves

    v8f c = {};

    // ---- K loop over two 64-row slabs of W staged in LDS ----
    for (int kb = 0; kb < DIM; kb += KSLAB) {
        __syncthreads();
        // load W[kb..kb+63][:] : 2048 float4s / 256 threads = 8 each
        {
            const float4* Wv  = (const float4*)W;
            float4*       sWv = (float4*)sW;
            #pragma unroll
            for (int i = 0; i < 8; ++i) {
                int idx = tid + 256 * i;                 // 0..2047
                sWv[idx] = Wv[kb * (DIM / 4) + idx];
            }
        }
        __syncthreads();

        #pragma unroll
        for (int k0 = 0; k0 < KSLAB; k0 += 4) {
            // A fragment: 16x4, lane holds row ln, K = k0+koff, k0+koff+1
            v2f a = *(const v2f*)&sH[ln * LDH + kb + k0 + koff];
            // B fragment: 4x16, lane holds col n0+ln, same K split
            v2f b;
            b.x = sW[(k0 + koff)     * DIM + n0 + ln];
            b.y = sW[(k0 + koff + 1) * DIM + n0 + ln];
            c = __builtin_amdgcn_wmma_f32_16x16x4_f32(
                    /*neg_a=*/false, a, /*neg_b=*/false, b,
                    /*c_mod=*/(short)0, c, /*reuse_a=*/false, /*reuse_b=*/false);
        }
    }

    // ---- epilogue: bias + leaky_relu + store ----
    // C/D layout: VGPR i -> lanes 0-15: M=i, N=lane; lanes 16-31: M=8+i, N=lane-16
    const float bv    = bias[n0 + ln];
    const int   mbase = (lane < 16) ? 0 : 8;
    #pragma unroll
    for (int i = 0; i < 8; ++i) {
        int r = row0 + mbase + i;
        if (r < n_nodes) {
            float v = c[i] + bv;
            v = (v > 0.0f) ? v : 0.01f * v;
            out[(size_t)r * DIM + n0 + ln] = v;
        }
    }
}

// ---------------------------------------------------------------------------
extern "C" void kernel_launch(void* const* d_in, const int* in_sizes, int n_in,
                              void* d_out, int out_size, void* d_ws, size_t ws_size,
                              hipStream_t stream) {
    const float* feat = (const float*)d_in[0];
    const int*   ei   = (const int*)d_in[1];   // edge_index, [2][E]
    const float* W1   = (const float*)d_in[2];
    const float* b1   = (const float*)d_in[3];
    const float* W2   = (const float*)d_in[4];
    const float* b2   = (const float*)d_in[5];
    float*       out  = (float*)d_out;

    const int N = in_sizes[0] / DIM;
    const int E = in_sizes[1] / 2;
    const int* src = ei;
    const int* dst = ei + E;

    float* acc = (float*)d_ws;                 // N*DIM accumulator
    float* h1  = acc + (size_t)N * DIM;        // N*DIM layer-1 output
    float* deg = h1  + (size_t)N * DIM;        // N degrees

    const int feat_elems = N * DIM;
    const int zgrid_feat = (feat_elems + 255) / 256;
    const int zgrid_deg  = (N + 255) / 256;
    const int egrid      = (E + 255) / 256;            // 1 thread / edge
    const int sgrid      = (E + 7) / 8;                // 1 wave / edge, 8 waves/block
    const int ggrid      = (N + TILE_M - 1) / TILE_M;  // 16 rows / block

    // degree (once) + layer-1 aggregation
    zero_f32_kernel<<<zgrid_feat, 256, 0, stream>>>(acc, feat_elems);
    zero_f32_kernel<<<zgrid_deg, 256, 0, stream>>>(deg, N);
    degree_kernel<<<egrid, 256, 0, stream>>>(dst, deg, E);
    scatter_add_kernel<<<sgrid, 256, 0, stream>>>(feat, src, dst, acc, E);

    // layer 1: h1 = leaky_relu(((acc + feat)/(deg+1)) @ W1 + b1)
    sage_gemm_wmma_kernel<<<ggrid, 256, 0, stream>>>(acc, feat, deg, W1, b1, h1, N);

    // layer-2 aggregation
    zero_f32_kernel<<<zgrid_feat, 256, 0, stream>>>(acc, feat_elems);
    scatter_add_kernel<<<sgrid, 256, 0, stream>>>(h1, src, dst, acc, E);

    // layer 2: out = leaky_relu(((acc + h1)/(deg+1)) @ W2 + b2)
    sage_gemm_wmma_kernel<<<ggrid, 256, 0, stream>>>(acc, h1, deg, W2, b2, out, N);
}